// SpectralCompetitionScheduler_70944269795693
// MI455X (gfx1250) — compile-verified
//
#include <hip/hip_runtime.h>
#include <math.h>

// Problem constants (match reference)
#define V    1024
#define NH   16384
#define NM   16384
#define OUTW (NH + NM)

typedef float v2f __attribute__((ext_vector_type(2)));
typedef float v4f __attribute__((ext_vector_type(4)));
typedef float v8f __attribute__((ext_vector_type(8)));

// ---------------- Pass 0: zero the atomic accumulators -------------------
__global__ void k0_zero(float* __restrict__ eRaw) {
    int i = blockIdx.x * blockDim.x + threadIdx.x;
    if (i < 3 * V) eRaw[i] = 0.0f;
}

// ---------------- Pass 1: per-voice per-band energy via WMMA -------------
// Each wave (block of 32 threads) owns a 16-voice tile and a K-slice.
// A-fragment (16x4 f32): lane L holds row (L&15); half = L>>4 selects K slots.
// B = all-ones => D rows replicate the band-masked row sums.
#define KSPLIT 64
#define KPER   (NH / KSPLIT)   // 256 harmonics per wave

__global__ void k1_band_energy(const float* __restrict__ harm,
                               const float* __restrict__ f0,
                               float* __restrict__ eRaw) {
    const int lane = threadIdx.x;            // 0..31 (wave32)
    const int tile = blockIdx.x >> 6;        // / KSPLIT
    const int ksl  = blockIdx.x & (KSPLIT - 1);
    const int row  = lane & 15;
    const int half = lane >> 4;              // 0 or 1
    const int gRow = tile * 16 + row;

    const float f0v = f0[gRow];
    const float* rp = harm + (size_t)gRow * NH;

    v8f cL = {}; v8f cM = {}; v8f cH = {};
    v2f ones = {1.0f, 1.0f};

    const int kBeg = ksl * KPER;
    for (int k = kBeg; k < kBeg + KPER; k += 8) {
        // b128 load: 4 consecutive floats of this voice's row.
        const int kb = k + half * 4;
        const v4f a4 = *(const v4f*)(rp + kb);

        // Band of each element: harmonics are 1-based -> freq = f0 * (kb+1+e)
        const float fr0 = f0v * (float)(kb + 1);
        const float fr1 = f0v * (float)(kb + 2);
        const float fr2 = f0v * (float)(kb + 3);
        const float fr3 = f0v * (float)(kb + 4);
        const int b0 = (fr0 >= 500.0f) + (fr0 >= 2000.0f);
        const int b1 = (fr1 >= 500.0f) + (fr1 >= 2000.0f);
        const int b2 = (fr2 >= 500.0f) + (fr2 >= 2000.0f);
        const int b3 = (fr3 >= 500.0f) + (fr3 >= 2000.0f);

        // First K-chunk: elements 0,1 of the float4
        v2f aL = { (b0 == 0) ? a4.x : 0.0f, (b1 == 0) ? a4.y : 0.0f };
        v2f aM = { (b0 == 1) ? a4.x : 0.0f, (b1 == 1) ? a4.y : 0.0f };
        v2f aH = { (b0 == 2) ? a4.x : 0.0f, (b1 == 2) ? a4.y : 0.0f };
        cL = __builtin_amdgcn_wmma_f32_16x16x4_f32(false, aL, false, ones, (short)0, cL, false, false);
        cM = __builtin_amdgcn_wmma_f32_16x16x4_f32(false, aM, false, ones, (short)0, cM, false, false);
        cH = __builtin_amdgcn_wmma_f32_16x16x4_f32(false, aH, false, ones, (short)0, cH, false, false);

        // Second K-chunk: elements 2,3
        v2f bL = { (b2 == 0) ? a4.z : 0.0f, (b3 == 0) ? a4.w : 0.0f };
        v2f bM = { (b2 == 1) ? a4.z : 0.0f, (b3 == 1) ? a4.w : 0.0f };
        v2f bH = { (b2 == 2) ? a4.z : 0.0f, (b3 == 2) ? a4.w : 0.0f };
        cL = __builtin_amdgcn_wmma_f32_16x16x4_f32(false, bL, false, ones, (short)0, cL, false, false);
        cM = __builtin_amdgcn_wmma_f32_16x16x4_f32(false, bM, false, ones, (short)0, cM, false, false);
        cH = __builtin_amdgcn_wmma_f32_16x16x4_f32(false, bH, false, ones, (short)0, cH, false, false);
    }

    // D layout: VGPR r, lanes 0-15 -> M=r (N=lane), lanes 16-31 -> M=r+8.
    // Column 0 lives in lanes 0 and 16; those lanes publish the row sums.
    if ((lane & 15) == 0) {
#pragma unroll
        for (int r = 0; r < 8; ++r) {
            const int m = tile * 16 + half * 8 + r;
            atomicAdd(&eRaw[m * 3 + 0], cL[r]);
            atomicAdd(&eRaw[m * 3 + 1], cM[r]);
            atomicAdd(&eRaw[m * 3 + 2], cH[r]);
        }
    }
}

// ---------------- Pass 2: competition math (one 1024-thread block) -------
__global__ void k2_gains(const float* __restrict__ eRaw,
                         const float* __restrict__ cw,
                         const float* __restrict__ wd,
                         const unsigned char* __restrict__ act,
                         float* __restrict__ gEff,
                         float* __restrict__ nEff) {
    const int v = threadIdx.x;              // one voice per thread, V == blockDim
    const float a   = act[v] ? 1.0f : 0.0f;
    const float e0  = eRaw[v * 3 + 0];
    const float e1  = eRaw[v * 3 + 1];
    const float e2  = eRaw[v * 3 + 2];
    const float cwv = cw[v];

    // Reduce 5 quantities across the block: tE[3], total_weight, n_active
    float r0 = e0 * a, r1 = e1 * a, r2 = e2 * a, r3 = cwv * a, r4 = a;
    __shared__ float s[5][32];
    const int lane = v & 31, wid = v >> 5;
#pragma unroll
    for (int off = 16; off > 0; off >>= 1) {
        r0 += __shfl_down(r0, off, 32);
        r1 += __shfl_down(r1, off, 32);
        r2 += __shfl_down(r2, off, 32);
        r3 += __shfl_down(r3, off, 32);
        r4 += __shfl_down(r4, off, 32);
    }
    if (lane == 0) { s[0][wid] = r0; s[1][wid] = r1; s[2][wid] = r2; s[3][wid] = r3; s[4][wid] = r4; }
    __syncthreads();
    if (wid == 0) {
        float q0 = s[0][lane], q1 = s[1][lane], q2 = s[2][lane], q3 = s[3][lane], q4 = s[4][lane];
#pragma unroll
        for (int off = 16; off > 0; off >>= 1) {
            q0 += __shfl_down(q0, off, 32);
            q1 += __shfl_down(q1, off, 32);
            q2 += __shfl_down(q2, off, 32);
            q3 += __shfl_down(q3, off, 32);
            q4 += __shfl_down(q4, off, 32);
        }
        if (lane == 0) { s[0][0] = q0; s[1][0] = q1; s[2][0] = q2; s[3][0] = q3; s[4][0] = q4; }
    }
    __syncthreads();
    const float tE[3]  = { s[0][0], s[1][0], s[2][0] };
    const float tW     = s[3][0];
    const float nA     = s[4][0];
    const bool changed = (nA >= 2.0f);

    const float share = (cwv / fmaxf(tW, 1e-6f)) * 0.5f;   // claim_ratio * ENERGY_THRESHOLD
    const float eA[3] = { e0 * a, e1 * a, e2 * a };

    float g[3];
#pragma unroll
    for (int b = 0; b < 3; ++b) {
        const float excess = eA[b] - share;
        const float er     = excess / fmaxf(eA[b], 1e-6f);
        const float red    = fmaxf(0.3f, 1.0f - wd[v * 3 + b] * er * 0.5f);
        const bool apply   = (tE[b] > 0.5f) && (eA[b] > 0.0f) && (excess > 0.0f) && (a > 0.0f);
        g[b] = apply ? red : 1.0f;
    }

    // Bands partition the harmonics, so the row totals are band-sum combos.
    const float tb = e0 + e1 + e2;
    const float ta = g[0] * e0 + g[1] * e1 + g[2] * e2;
    const float ns = (tb > 1e-6f) ? (ta / tb) : 1.0f;

    gEff[v * 3 + 0] = changed ? g[0] : 1.0f;   // inactive voices already have g==1
    gEff[v * 3 + 1] = changed ? g[1] : 1.0f;
    gEff[v * 3 + 2] = changed ? g[2] : 1.0f;
    nEff[v]         = (changed && a > 0.0f) ? ns : 1.0f;
}

// ---------------- Pass 3: streaming apply (float4, NT output) ------------
__global__ void k3_apply(const float* __restrict__ harm,
                         const float* __restrict__ noise,
                         const float* __restrict__ f0,
                         const float* __restrict__ gEff,
                         const float* __restrict__ nEff,
                         float* __restrict__ out) {
    const int idx = blockIdx.x * blockDim.x + threadIdx.x;  // float4 index
    const int row = idx >> 13;                              // OUTW/4 == 8192 float4 per row
    const int c4  = idx & 8191;

    if (c4 < (NH / 4)) {
        const int k = c4 * 4;
        // harm was read in pass 1 and should still be L2-resident: regular load
        const v4f a = *(const v4f*)(harm + (size_t)row * NH + k);
        const float f0v = f0[row];
        const float g0 = gEff[row * 3 + 0];
        const float g1 = gEff[row * 3 + 1];
        const float g2 = gEff[row * 3 + 2];
        v4f o;
#pragma unroll
        for (int e = 0; e < 4; ++e) {
            const float fr = f0v * (float)(k + 1 + e);
            const float g  = (fr < 500.0f) ? g0 : ((fr < 2000.0f) ? g1 : g2);
            o[e] = a[e] * g;
        }
        __builtin_nontemporal_store(o, (v4f*)(out + (size_t)row * OUTW + k));
    } else {
        const int j = (c4 - NH / 4) * 4;
        // noise is read exactly once: non-temporal to keep harm in L2
        const v4f n4 = __builtin_nontemporal_load((const v4f*)(noise + (size_t)row * NM + j));
        const float sc = nEff[row];
        v4f o = { n4.x * sc, n4.y * sc, n4.z * sc, n4.w * sc };
        __builtin_nontemporal_store(o, (v4f*)(out + (size_t)row * OUTW + NH + j));
    }
}

// ---------------- Launch --------------------------------------------------
extern "C" void kernel_launch(void* const* d_in, const int* in_sizes, int n_in,
                              void* d_out, int out_size, void* d_ws, size_t ws_size,
                              hipStream_t stream) {
    (void)in_sizes; (void)n_in; (void)out_size; (void)ws_size;

    const float*         harm  = (const float*)d_in[0];
    const float*         noise = (const float*)d_in[1];
    const float*         f0    = (const float*)d_in[2];
    const float*         cw    = (const float*)d_in[3];
    const float*         wd    = (const float*)d_in[4];
    const unsigned char* act   = (const unsigned char*)d_in[5];  // numpy bool: 1 byte/elem
    float* out = (float*)d_out;

    // Workspace: [0,3V) raw band energies (atomic acc), [3V,6V) eff gains, [6V,7V) noise mults
    float* eRaw = (float*)d_ws;
    float* gEff = eRaw + 3 * V;
    float* nEff = gEff + 3 * V;

    k0_zero<<<(3 * V + 255) / 256, 256, 0, stream>>>(eRaw);

    // 64 voice-tiles x KSPLIT K-slices, one wave per block
    k1_band_energy<<<64 * KSPLIT, 32, 0, stream>>>(harm, f0, eRaw);

    k2_gains<<<1, V, 0, stream>>>(eRaw, cw, wd, act, gEff, nEff);

    const int tot4 = V * (OUTW / 4);   // 8,388,608 float4 elements
    k3_apply<<<tot4 / 256, 256, 0, stream>>>(harm, noise, f0, gEff, nEff, out);
}